// smoothness_loss_29575144801114
// MI455X (gfx1250) — compile-verified
//
#include <hip/hip_runtime.h>
#include <stdint.h>

#define BATCH  64
#define IMH    512
#define IMW    512
#define TILE_H 8
#define TROWS  (TILE_H + 2)          // 10 rows per image incl. vertical halo
#define NBLK   (BATCH * (IMH / TILE_H))   // 4096

// ---------------------------------------------------------------------------
// Fused Sobel-x/Sobel-y/Laplace stencil + charbonnier loss, tile-partial sums.
// Data movement: CDNA5 async global->LDS (ASYNCcnt) for the 40KB row tile.
// ---------------------------------------------------------------------------
__global__ __launch_bounds__(512)
void sm_loss_stencil(const float* __restrict__ pred,
                     const float* __restrict__ targ,
                     float* __restrict__ partial)
{
    __shared__ __align__(16) float smem[2 * TROWS * IMW];   // 40 KB
    const int tid  = threadIdx.x;
    const int bimg = blockIdx.x >> 6;     // image index (64 row-tiles/image)
    const int ty   = blockIdx.x & 63;     // row-tile index
    const int y0   = ty * TILE_H;

    // LDS byte offset of smem base (low 32 bits of generic address == LDS addr)
    const uint32_t smbase = (uint32_t)(uintptr_t)&smem[0];

    // ---- stage 2*TROWS rows of IMW floats into LDS via async copies ----
    const int chunk = tid & 127;          // 128 x 16B chunks per 512-float row
    const int rslot = tid >> 7;           // 4 rows loaded per iteration
    for (int i = 0; i < 5; ++i) {
        const int r   = i * 4 + rslot;    // 0..19
        const int img = r / TROWS;        // 0 = pred, 1 = target
        const int rt  = r % TROWS;        // row within tile
        const int gy  = y0 - 1 + rt;      // global image row
        const uint32_t ldsaddr =
            smbase + (uint32_t)(((img * TROWS + rt) * IMW + chunk * 4) * 4);
        if ((unsigned)gy < (unsigned)IMH) {
            const float* src = img ? targ : pred;
            const uint64_t ga = (uint64_t)(uintptr_t)
                (src + ((size_t)bimg * IMH + (size_t)gy) * IMW + chunk * 4);
            asm volatile("global_load_async_to_lds_b128 %0, %1, off"
                         :: "v"(ldsaddr), "v"(ga) : "memory");
        } else {
            // zero-fill vertical halo rows outside the image (zero padding)
            float4 z = make_float4(0.f, 0.f, 0.f, 0.f);
            *(float4*)(&smem[(img * TROWS + rt) * IMW + chunk * 4]) = z;
        }
    }
    asm volatile("s_wait_asynccnt 0x0" ::: "memory");
    __syncthreads();

    // ---- compute: one column per thread, sliding 3x3 register window ----
    const int   x  = tid;
    const float* tp = &smem[0];
    const float* tq = &smem[TROWS * IMW];
    const int   xl = (x > 0)       ? x - 1 : 0;
    const int   xr = (x < IMW - 1) ? x + 1 : x;
    const float lm = (x > 0)       ? 1.f : 0.f;   // left-edge zero pad
    const float rm = (x < IMW - 1) ? 1.f : 0.f;   // right-edge zero pad

    float p00,p01,p02,p10,p11,p12,p20,p21,p22;
    float q00,q01,q02,q10,q11,q12,q20,q21,q22;

    p00 = tp[0*IMW+xl]*lm; p01 = tp[0*IMW+x]; p02 = tp[0*IMW+xr]*rm;
    p10 = tp[1*IMW+xl]*lm; p11 = tp[1*IMW+x]; p12 = tp[1*IMW+xr]*rm;
    q00 = tq[0*IMW+xl]*lm; q01 = tq[0*IMW+x]; q02 = tq[0*IMW+xr]*rm;
    q10 = tq[1*IMW+xl]*lm; q11 = tq[1*IMW+x]; q12 = tq[1*IMW+xr]*rm;

    float acc = 0.f;
    #pragma unroll
    for (int yy = 0; yy < TILE_H; ++yy) {
        const int rb = (yy + 2) * IMW;
        p20 = tp[rb+xl]*lm; p21 = tp[rb+x]; p22 = tp[rb+xr]*rm;
        q20 = tq[rb+xl]*lm; q21 = tq[rb+x]; q22 = tq[rb+xr]*rm;

        // Sobel-x, Sobel-y, Laplace on pred
        const float sxp = (p02 - p00) + 2.f*(p12 - p10) + (p22 - p20);
        const float syp = (p20 + 2.f*p21 + p22) - (p00 + 2.f*p01 + p02);
        const float lpp = 8.f*p11 - (p00+p01+p02+p10+p12+p20+p21+p22);
        // ... and on target
        const float sxq = (q02 - q00) + 2.f*(q12 - q10) + (q22 - q20);
        const float syq = (q20 + 2.f*q21 + q22) - (q00 + 2.f*q01 + q02);
        const float lpq = 8.f*q11 - (q00+q01+q02+q10+q12+q20+q21+q22);

        const float a1 = fabsf(sxp) * __expf(-10.f * fabsf(sxq));
        const float a2 = fabsf(syp) * __expf(-10.f * fabsf(syq));
        const float a3 = fabsf(lpp) * __expf(-10.f * fabsf(lpq));

        const float c1 = sqrtf(fmaf(a1, a1, 1.0e-6f));   // charbonnier, eps=1e-3
        const float c2 = sqrtf(fmaf(a2, a2, 1.0e-6f));
        const float c3 = sqrtf(fmaf(a3, a3, 1.0e-6f));

        acc += 10.f * (c1 + c2) + c3;                    // s1=10, s2=1

        p00=p10; p01=p11; p02=p12; p10=p20; p11=p21; p12=p22;
        q00=q10; q01=q11; q02=q12; q10=q20; q11=q21; q12=q22;
    }

    // ---- deterministic block reduction (reuse LDS) ----
    __syncthreads();
    float* red = (float*)smem;
    red[tid] = acc;
    __syncthreads();
    #pragma unroll
    for (int s = 256; s > 0; s >>= 1) {
        if (tid < s) red[tid] += red[tid + s];
        __syncthreads();
    }
    if (tid == 0) partial[blockIdx.x] = red[0];
}

// ---------------------------------------------------------------------------
// Final reduction of 4096 block partials (double accumulation), mean scale.
// ---------------------------------------------------------------------------
__global__ __launch_bounds__(512)
void sm_loss_reduce(const float* __restrict__ partial, float* __restrict__ out,
                    int n)
{
    __shared__ double red[512];
    double s = 0.0;
    for (int i = threadIdx.x; i < n; i += 512) s += (double)partial[i];
    red[threadIdx.x] = s;
    __syncthreads();
    #pragma unroll
    for (int t = 256; t > 0; t >>= 1) {
        if (threadIdx.x < t) red[threadIdx.x] += red[threadIdx.x + t];
        __syncthreads();
    }
    if (threadIdx.x == 0)
        out[0] = (float)(red[0] * (1.0 / ((double)BATCH * IMH * IMW)));
}

extern "C" void kernel_launch(void* const* d_in, const int* in_sizes, int n_in,
                              void* d_out, int out_size, void* d_ws, size_t ws_size,
                              hipStream_t stream)
{
    (void)in_sizes; (void)n_in; (void)out_size; (void)ws_size;
    const float* pred = (const float*)d_in[0];
    const float* targ = (const float*)d_in[1];
    float* out     = (float*)d_out;
    float* partial = (float*)d_ws;          // NBLK floats = 16 KB scratch

    sm_loss_stencil<<<dim3(NBLK), dim3(512), 0, stream>>>(pred, targ, partial);
    sm_loss_reduce <<<dim3(1),    dim3(512), 0, stream>>>(partial, out, NBLK);
}